// FeedForwardQuantum_65481071400590
// MI455X (gfx1250) — compile-verified
//
#include <hip/hip_runtime.h>

typedef __attribute__((ext_vector_type(2))) float v2f;
typedef __attribute__((ext_vector_type(8))) float v8f;
typedef int b128v __attribute__((vector_size(16)));   // matches builtin's V4i param

#define TOKENS 16384
#define EMBED  512
#define FFN    2048
#define NQ     8
#define MT     64            // tokens per workgroup
#define KC     8             // ffn-chunk staged per iteration (double-buffered)
#define LDSTR  (KC + 4)      // padded LDS row stride: 48B rows, 16B aligned, conflict-free
#define NCH    (FFN / KC)

#if defined(__has_builtin)
#if __has_builtin(__builtin_amdgcn_global_load_async_to_lds_b128)
#define HAVE_ASYNC_LDS 1
#endif
#endif
#ifndef HAVE_ASYNC_LDS
#define HAVE_ASYNC_LDS 0
#endif

__device__ __forceinline__ void wait_async_lds() {
#if HAVE_ASYNC_LDS
#if __has_builtin(__builtin_amdgcn_s_wait_asynccnt)
    __builtin_amdgcn_s_wait_asynccnt(0);
#else
    asm volatile("s_wait_asynccnt 0x0" ::: "memory");
#endif
#endif
}

// Stage W2[e][kc:kc+KC] for all 512 e into an LDS buffer.
// Async path: global_load_async_to_lds_b128 (ASYNCcnt), 4 issues/thread.
__device__ __forceinline__ void stage_w2(const float* __restrict__ W2, int kc,
                                         float (*__restrict__ w2buf)[LDSTR], int tid) {
    const int r  = tid >> 1;          // 0..127
    const int c4 = (tid & 1) * 4;     // 0 or 4 floats
#pragma unroll
    for (int p = 0; p < EMBED; p += 128) {
        const float* g = &W2[(size_t)(r + p) * FFN + kc + c4];
        float*       l = &w2buf[r + p][c4];
#if HAVE_ASYNC_LDS
        __builtin_amdgcn_global_load_async_to_lds_b128(
            (__attribute__((address_space(1))) b128v*)g,
            (__attribute__((address_space(3))) b128v*)l,
            0, 0);
#else
        const float4 v = *(const float4*)g;
        *(float4*)l = v;
#endif
    }
}

// h[t][f] = relu( sum_q z[t][q] * W1[f][q] ), f in [kc, kc+KC)
__device__ __forceinline__ void compute_h(const float* __restrict__ W1, int kc,
                                          const float (*__restrict__ zt)[NQ],
                                          float (*__restrict__ hbuf)[LDSTR], int tid) {
    const int f  = kc + (tid & 7);
    const int tb = (tid >> 3) * 2;    // 2 tokens per thread
    const float4 w1a = *(const float4*)&W1[f * NQ];
    const float4 w1b = *(const float4*)&W1[f * NQ + 4];
#pragma unroll
    for (int i = 0; i < 2; ++i) {
        const int t = tb + i;
        float s = zt[t][0]*w1a.x + zt[t][1]*w1a.y + zt[t][2]*w1a.z + zt[t][3]*w1a.w
                + zt[t][4]*w1b.x + zt[t][5]*w1b.y + zt[t][6]*w1b.z + zt[t][7]*w1b.w;
        hbuf[t][tid & 7] = s > 0.0f ? s : 0.0f;
    }
}

__launch_bounds__(256)
__global__ void ffq_fused_kernel(const float* __restrict__ x,
                                 const float* __restrict__ theta,
                                 const float* __restrict__ W1,
                                 const float* __restrict__ W2,
                                 float* __restrict__ out) {
    __shared__ float ct[NQ];
    __shared__ float zt[MT][NQ];
    __shared__ float hc[2][MT][LDSTR];      // h chunks (A), double-buffered
    __shared__ float w2c[2][EMBED][LDSTR];  // W2 chunks (B, [n][k]), double-buffered

    const int tid    = threadIdx.x;
    const int lane   = tid & 31;
    const int lane15 = lane & 15;
    const int laneHi = lane >> 4;
    const int wave   = tid >> 5;
    const int t0     = blockIdx.x * MT;

    // Kick off chunk-0 W2 staging immediately (overlaps with the encoder below).
    stage_w2(W2, 0, w2c[0], tid);
    if (tid < NQ) ct[tid] = __builtin_cosf(theta[tid]);
    __syncthreads();   // ct visible

    // Quantum encoder: z[t][q] = cos(x[t][q]) * cos(theta[q])
    for (int i = tid; i < MT * NQ; i += 256) {
        const int t = i >> 3, q = i & 7;
        zt[t][q] = __builtin_cosf(x[(size_t)(t0 + t) * EMBED + q]) * ct[q];
    }
    wait_async_lds();
    __syncthreads();   // zt + w2c[0] visible

    compute_h(W1, 0, zt, hc[0], tid);
    __syncthreads();   // hc[0] visible

    // Wave tiling: 2 M-subtiles x 8 N-subtiles of 16x16 per wave
    const int mW = (wave >> 2) * 32;
    const int nW = (wave & 3) * 128;

    v8f acc[2][8];
#pragma unroll
    for (int i = 0; i < 2; ++i)
#pragma unroll
        for (int j = 0; j < 8; ++j)
            acc[i][j] = (v8f){};

    for (int c = 0; c < NCH; ++c) {
        const int buf = c & 1;

        // Pipeline: stage chunk c+1 (async DMA + VALU h) while WMMA eats chunk c.
        if (c + 1 < NCH) {
            stage_w2(W2, (c + 1) * KC, w2c[buf ^ 1], tid);
            compute_h(W1, (c + 1) * KC, zt, hc[buf ^ 1], tid);
        }

        // FP32 WMMA: D = A(16x4) * B(4x16) + C over this chunk.
        // A frag (ISA 7.12.2): lanes 0-15 -> M=lane, K={kk,kk+1}; lanes 16-31 -> K={kk+2,kk+3}.
#pragma unroll
        for (int kk = 0; kk < KC; kk += 4) {
            const int ko = kk + laneHi * 2;
            const v2f a0 = *(const v2f*)&hc[buf][mW +      lane15][ko];
            const v2f a1 = *(const v2f*)&hc[buf][mW + 16 + lane15][ko];
#pragma unroll
            for (int nt = 0; nt < 8; ++nt) {
                const v2f b = *(const v2f*)&w2c[buf][nW + nt * 16 + lane15][ko];
                acc[0][nt] = __builtin_amdgcn_wmma_f32_16x16x4_f32(
                    false, a0, false, b, (short)0, acc[0][nt], false, false);
                acc[1][nt] = __builtin_amdgcn_wmma_f32_16x16x4_f32(
                    false, a1, false, b, (short)0, acc[1][nt], false, false);
            }
        }

        wait_async_lds();   // our async W2 writes for chunk c+1 landed in LDS
        __syncthreads();    // all waves done reading buf; hc/w2c[buf^1] visible
    }

    // Store C/D tiles: VGPR j -> row M=j (lanes 0-15) / M=j+8 (lanes 16-31), col N=lane15.
#pragma unroll
    for (int mi = 0; mi < 2; ++mi) {
#pragma unroll
        for (int nt = 0; nt < 8; ++nt) {
            const int col     = nW + nt * 16 + lane15;
            const int rowbase = t0 + mW + mi * 16 + laneHi * 8;
#pragma unroll
            for (int j = 0; j < 8; ++j) {
                out[(size_t)(rowbase + j) * EMBED + col] = acc[mi][nt][j];
            }
        }
    }
}

extern "C" void kernel_launch(void* const* d_in, const int* in_sizes, int n_in,
                              void* d_out, int out_size, void* d_ws, size_t ws_size,
                              hipStream_t stream) {
    const float* x     = (const float*)d_in[0];
    const float* theta = (const float*)d_in[1];
    const float* W1    = (const float*)d_in[2];
    const float* W2    = (const float*)d_in[3];
    float* out         = (float*)d_out;
    (void)in_sizes; (void)n_in; (void)out_size; (void)d_ws; (void)ws_size;

    dim3 grid(TOKENS / MT);   // 256 workgroups
    dim3 block(256);          // 8 wave32s
    ffq_fused_kernel<<<grid, block, 0, stream>>>(x, theta, W1, W2, out);
}